// ContextRNN_86354612453585
// MI455X (gfx1250) — compile-verified
//
#include <hip/hip_runtime.h>
#include <math.h>

// ContextRNN (bidirectional GRU) for MI455X / gfx1250, wave32, fp32 WMMA.
//
// Sizes (fixed by reference): B=32, T=512, M=4, V=32000, H=512, 3H=1536.
//
// Pipeline:
//   0) memset: hidden-state double buffers + grid-barrier counters = 0
//   1) embed:   embedded[b][t][h] = sum_m emb[seq[b][t][m]][h]
//   2) swizzle: Wx_f/Wx_b/Wh_f/Wh_b/Wd -> WMMA B-fragment layout
//   3) gemm:    Gx_d[t*32+b][3H] = embedded[b][t] @ Wx_d + b_d[0]   (per dir)
//   4) scan:    persistent kernel, 32 blocks (2 dirs x 16 hidden slices),
//               per step: 12 waves x 128 v_wmma_f32_16x16x4_f32 -> gh tile,
//               elementwise GRU update, per-direction grid barrier.
//   5) gemm:    outputs = [out_f|out_b] @ Wd + bd   (K split across two A's)
//   6) gemm:    hidden  = [h_f|h_b] @ Wd + bd

typedef float v2f __attribute__((ext_vector_type(2)));
typedef float v8f __attribute__((ext_vector_type(8)));

#define Bsz 32
#define Tsz 512
#define Hsz 512
#define H3  1536

// ---------------------------------------------------------------------------
// WMMA fragment layout helpers (CDNA5 ISA 7.12.2, 32-bit 16x16x4):
//  A (16x4, MxK): lane L: M=L&15, regs r=0,1 hold K = 2*(L>>4)+r
//  B (4x16, KxN): lane L: N=L&15, regs r=0,1 hold K = 2*(L>>4)+r
//  C/D (16x16):   lane L: N=L&15, reg v holds M = v + 8*(L>>4)
// Swizzled weight layout: Sw[ntile][kstep][lane][r], so a wave's per-kstep
// fragment load is 32 consecutive float2 (256B, fully coalesced).
// ---------------------------------------------------------------------------

__global__ __launch_bounds__(256) void embed_kernel(
    const int* __restrict__ seqs, const float* __restrict__ emb,
    float* __restrict__ outE)
{
  const long bt = blockIdx.x;                    // = b*T + t
  const int i0 = seqs[bt * 4 + 0];
  const int i1 = seqs[bt * 4 + 1];
  const int i2 = seqs[bt * 4 + 2];
  const int i3 = seqs[bt * 4 + 3];
  for (int h = threadIdx.x; h < Hsz; h += 256) {
    outE[bt * Hsz + h] = emb[(long)i0 * Hsz + h] + emb[(long)i1 * Hsz + h] +
                         emb[(long)i2 * Hsz + h] + emb[(long)i3 * Hsz + h];
  }
}

__global__ __launch_bounds__(256) void swizzleB_kernel(
    const float* __restrict__ W, float* __restrict__ Sw, int K, int N)
{
  const int Kstep = K >> 2;
  const long total = (long)K * N;
  for (long e = (long)blockIdx.x * blockDim.x + threadIdx.x; e < total;
       e += (long)gridDim.x * blockDim.x) {
    const int r    = (int)(e & 1);
    const int lane = (int)((e >> 1) & 31);
    const long ch  = e >> 6;
    const int ks   = (int)(ch % Kstep);
    const int nt   = (int)(ch / Kstep);
    const int k = 4 * ks + 2 * (lane >> 4) + r;
    const int n = nt * 16 + (lane & 15);
    Sw[e] = W[(long)k * N + n];
  }
}

// Generic fp32 WMMA GEMM: C[r][n] = A[rowmap(r)] . Bsw[:,n] + bias[n]
// A may be a concatenation of A1 (K1steps*4 cols) and A2 (K2steps*4 cols).
// amode: 0 -> A row = r (stride lda);  1 -> A row offset = ((r&31)*T + (r>>5))*lda
// cmode: 0 -> C[r*ldc+n];              1 -> C[((r&31)*T + (r>>5))*ldc + n]
__global__ __launch_bounds__(256) void gemm_wmma(
    const float* __restrict__ A1, const float* __restrict__ A2,
    const float* __restrict__ Bsw, const float* __restrict__ bias,
    float* __restrict__ C,
    int Mrows, int Ntiles, int K1steps, int K2steps,
    int lda, int ldc, int amode, int cmode)
{
  const int lane = threadIdx.x & 31;
  const int wave = threadIdx.x >> 5;
  const int MT = Mrows >> 4;
  const long total = (long)MT * Ntiles;
  const long tile = (long)blockIdx.x * 8 + wave;
  if (tile >= total) return;                     // wave-uniform: EXEC stays full
  const int nt = (int)(tile % Ntiles);
  const int mt = (int)(tile / Ntiles);

  const int rowA = mt * 16 + (lane & 15);
  const int klo  = 2 * (lane >> 4);
  long aoff;
  if (amode == 1) aoff = ((long)(rowA & 31) * Tsz + (rowA >> 5)) * lda;
  else            aoff = (long)rowA * lda;

  const int Kt = K1steps + K2steps;
  const float* bptr  = Bsw + (long)nt * Kt * 64 + lane * 2;
  const float* aptr1 = A1 + aoff + klo;

  v8f acc = {};
  for (int ks = 0; ks < K1steps; ++ks) {
    v2f a = *(const v2f*)(aptr1 + 4 * ks);
    v2f b = *(const v2f*)(bptr + (long)ks * 64);
    acc = __builtin_amdgcn_wmma_f32_16x16x4_f32(false, a, false, b,
                                                (short)0, acc, false, false);
  }
  if (K2steps > 0) {
    const float* aptr2 = A2 + aoff + klo;
    for (int ks = 0; ks < K2steps; ++ks) {
      v2f a = *(const v2f*)(aptr2 + 4 * ks);
      v2f b = *(const v2f*)(bptr + (long)(K1steps + ks) * 64);
      acc = __builtin_amdgcn_wmma_f32_16x16x4_f32(false, a, false, b,
                                                  (short)0, acc, false, false);
    }
  }

  const int col = nt * 16 + (lane & 15);
  const float bv = bias ? bias[col] : 0.0f;
#pragma unroll
  for (int v = 0; v < 8; ++v) {
    const int row = mt * 16 + v + 8 * (lane >> 4);
    long idx;
    if (cmode == 1) idx = ((long)(row & 31) * Tsz + (row >> 5)) * ldc + col;
    else            idx = (long)row * ldc + col;
    C[idx] = acc[v] + bv;
  }
}

// ---------------------------------------------------------------------------
// Persistent recurrent scan. grid = 32 blocks (blockIdx>>4 = direction,
// blockIdx&15 = 32-wide hidden slice), 384 threads = 12 waves.
// Per step: gh(32x96) = h(32x512) @ WhSlice via 12 one-tile waves (K=512),
// then elementwise GRU update, then a per-direction 16-block grid barrier.
// h is kept in global memory twice: plain [32][512] and WMMA-A-swizzled
// [16384] so the GEMM A-fragment loads are coalesced float2 reads.
// ---------------------------------------------------------------------------
__global__ __launch_bounds__(384) void gru_scan_kernel(
    const float* __restrict__ Gx0, const float* __restrict__ Gx1,
    const float* __restrict__ WhSw0, const float* __restrict__ WhSw1,
    const float* __restrict__ br0, const float* __restrict__ br1,
    float* __restrict__ Hplain,   // [2][2][32*512]
    float* __restrict__ Hsw,      // [2][2][16384]
    float* __restrict__ Out0, float* __restrict__ Out1,  // [T*32][512] each
    unsigned* __restrict__ bar)   // [2]{cnt,gen}
{
  __shared__ float ghL[32 * 96];

  const int tid  = threadIdx.x;
  const int lane = tid & 31;
  const int wave = tid >> 5;
  const int d = blockIdx.x >> 4;
  const int g = blockIdx.x & 15;

  const float* Gx   = d ? Gx1 : Gx0;
  const float* WhSw = d ? WhSw1 : WhSw0;
  const float* br   = d ? br1 : br0;
  float* Out = d ? Out1 : Out0;
  float* Hp = Hplain + (size_t)d * 2 * 16384;
  float* Hs = Hsw    + (size_t)d * 2 * 16384;
  unsigned* cnt = bar + (size_t)d * 2;
  unsigned* gen = bar + (size_t)d * 2 + 1;

  const int m  = wave & 1;       // M tile (batch half)
  const int nl = wave >> 1;      // 0..5 local N tile
  // local n tile -> global 16-col tile in the 1536-wide gh space (z|r|n blocks)
  const int ntile = (nl >> 1) * 64 + 2 * g + (nl & 1);
  const float* Bbase = WhSw + (size_t)ntile * 128 * 64 + lane * 2;

  for (int i = 0; i < Tsz; ++i) {
    const int p = i & 1;
    const int t = d ? (Tsz - 1 - i) : i;

    // ---- GEMM: one 16x16 tile per wave, K = 512 (128 k-steps) ----
    {
      const float* Abase = Hs + (size_t)p * 16384 + (size_t)m * 128 * 64 + lane * 2;
      v8f acc = {};
#pragma unroll 4
      for (int ks = 0; ks < 128; ++ks) {
        v2f a = *(const v2f*)(Abase + ks * 64);
        v2f b = *(const v2f*)(Bbase + ks * 64);
        acc = __builtin_amdgcn_wmma_f32_16x16x4_f32(false, a, false, b,
                                                    (short)0, acc, false, false);
      }
      const int colB = 16 * nl + (lane & 15);
      const int rowB = 16 * m + 8 * (lane >> 4);
#pragma unroll
      for (int v = 0; v < 8; ++v) ghL[(rowB + v) * 96 + colB] = acc[v];
    }
    __syncthreads();

    // ---- elementwise GRU update for this 32x32 hidden slice ----
    for (int e = tid; e < 1024; e += 384) {
      const int b = e >> 5, j = e & 31;
      const int col = g * 32 + j;
      const size_t rGx = (size_t)t * 32 + b;
      const float hz = ghL[b * 96 + j]      + br[col];
      const float hr = ghL[b * 96 + 32 + j] + br[512 + col];
      const float hn = ghL[b * 96 + 64 + j] + br[1024 + col];
      const float gxz = Gx[rGx * H3 + col];
      const float gxr = Gx[rGx * H3 + 512 + col];
      const float gxn = Gx[rGx * H3 + 1024 + col];
      const float hold = Hp[(size_t)p * 16384 + b * Hsz + col];
      const float z = 1.0f / (1.0f + expf(-(gxz + hz)));
      const float r = 1.0f / (1.0f + expf(-(gxr + hr)));
      const float hc = tanhf(gxn + r * hn);
      const float hnew = z * hold + (1.0f - z) * hc;
      Hp[(size_t)(1 - p) * 16384 + b * Hsz + col] = hnew;
      const int ks = col >> 2, kr = col & 3;
      const int L = ((kr >> 1) << 4) | (b & 15);
      Hs[(size_t)(1 - p) * 16384 + (((b >> 4) * 128 + ks) * 32 + L) * 2 + (kr & 1)] = hnew;
      Out[rGx * Hsz + col] = hnew;
    }

    // ---- per-direction grid barrier (16 blocks, sense-reversing) ----
    __threadfence();
    __syncthreads();
    if (tid == 0) {
      unsigned myg = __hip_atomic_load(gen, __ATOMIC_RELAXED, __HIP_MEMORY_SCOPE_AGENT);
      unsigned a = __hip_atomic_fetch_add(cnt, 1u, __ATOMIC_ACQ_REL, __HIP_MEMORY_SCOPE_AGENT);
      if (a == 15u) {
        __hip_atomic_store(cnt, 0u, __ATOMIC_RELAXED, __HIP_MEMORY_SCOPE_AGENT);
        __hip_atomic_fetch_add(gen, 1u, __ATOMIC_RELEASE, __HIP_MEMORY_SCOPE_AGENT);
      } else {
        while (__hip_atomic_load(gen, __ATOMIC_ACQUIRE, __HIP_MEMORY_SCOPE_AGENT) == myg)
          __builtin_amdgcn_s_sleep(2);
      }
    }
    __syncthreads();
    __threadfence();   // every thread: invalidate near caches before next read
  }
}

// ---------------------------------------------------------------------------
extern "C" void kernel_launch(void* const* d_in, const int* in_sizes, int n_in,
                              void* d_out, int out_size, void* d_ws, size_t ws_size,
                              hipStream_t stream)
{
  const int*   seqs = (const int*)d_in[0];
  // d_in[1] = input_lengths (unused by reference math)
  const float* emb  = (const float*)d_in[2];
  const float* Wx_f = (const float*)d_in[3];
  const float* Wh_f = (const float*)d_in[4];
  const float* b_f  = (const float*)d_in[5];   // (2,1536)
  const float* Wx_b = (const float*)d_in[6];
  const float* Wh_b = (const float*)d_in[7];
  const float* b_b  = (const float*)d_in[8];
  const float* Wd   = (const float*)d_in[9];   // (1024,512)
  const float* bd   = (const float*)d_in[10];
  float* outp = (float*)d_out;

  char* ws = (char*)d_ws;
  size_t cur = 0;
  auto alloc = [&](size_t bytes) {
    size_t o = cur;
    cur = (cur + bytes + 255) & ~(size_t)255;
    return o;
  };
  const size_t off_bar = alloc(4 * sizeof(unsigned));
  const size_t off_Hp  = alloc((size_t)2 * 2 * 16384 * 4);
  const size_t off_Hs  = alloc((size_t)2 * 2 * 16384 * 4);
  const size_t zero_end = cur;
  const size_t off_emb = alloc((size_t)16384 * Hsz * 4);
  const size_t off_Gx0 = alloc((size_t)16384 * H3 * 4);
  const size_t off_Gx1 = alloc((size_t)16384 * H3 * 4);
  const size_t off_O0  = alloc((size_t)16384 * Hsz * 4);
  const size_t off_O1  = alloc((size_t)16384 * Hsz * 4);
  const size_t off_WxS0 = alloc((size_t)Hsz * H3 * 4);
  const size_t off_WxS1 = alloc((size_t)Hsz * H3 * 4);
  const size_t off_WhS0 = alloc((size_t)Hsz * H3 * 4);
  const size_t off_WhS1 = alloc((size_t)Hsz * H3 * 4);
  const size_t off_WdS  = alloc((size_t)1024 * Hsz * 4);

  float* embd = (float*)(ws + off_emb);
  float* Gx0  = (float*)(ws + off_Gx0);
  float* Gx1  = (float*)(ws + off_Gx1);
  float* O0   = (float*)(ws + off_O0);
  float* O1   = (float*)(ws + off_O1);
  float* WxS0 = (float*)(ws + off_WxS0);
  float* WxS1 = (float*)(ws + off_WxS1);
  float* WhS0 = (float*)(ws + off_WhS0);
  float* WhS1 = (float*)(ws + off_WhS1);
  float* WdS  = (float*)(ws + off_WdS);
  float* Hp   = (float*)(ws + off_Hp);
  float* Hs   = (float*)(ws + off_Hs);
  unsigned* bar = (unsigned*)(ws + off_bar);

  // 0) zero h0 double-buffers (plain + swizzled) and barrier counters
  hipMemsetAsync(ws, 0, zero_end, stream);

  // 1) embedding gather + sum over M
  embed_kernel<<<Bsz * Tsz, 256, 0, stream>>>(seqs, emb, embd);

  // 2) swizzle weights into WMMA B-fragment layout
  swizzleB_kernel<<<1024, 256, 0, stream>>>(Wx_f, WxS0, Hsz, H3);
  swizzleB_kernel<<<1024, 256, 0, stream>>>(Wx_b, WxS1, Hsz, H3);
  swizzleB_kernel<<<1024, 256, 0, stream>>>(Wh_f, WhS0, Hsz, H3);
  swizzleB_kernel<<<1024, 256, 0, stream>>>(Wh_b, WhS1, Hsz, H3);
  swizzleB_kernel<<<1024, 256, 0, stream>>>(Wd,  WdS, 2 * Hsz, Hsz);

  // 3) Gx_d[t*32+b] = embedded[b][t] @ Wx_d + b_d[0]   (M=16384, N=1536, K=512)
  {
    const int blocks = (1024 * 96 + 7) / 8;   // 12288
    gemm_wmma<<<blocks, 256, 0, stream>>>(embd, nullptr, WxS0, b_f, Gx0,
                                          16384, 96, 128, 0, Hsz, H3, 1, 0);
    gemm_wmma<<<blocks, 256, 0, stream>>>(embd, nullptr, WxS1, b_b, Gx1,
                                          16384, 96, 128, 0, Hsz, H3, 1, 0);
  }

  // 4) persistent recurrent scan (both directions)
  gru_scan_kernel<<<32, 384, 0, stream>>>(Gx0, Gx1, WhS0, WhS1,
                                          b_f + H3, b_b + H3,
                                          Hp, Hs, O0, O1, bar);

  // 5) outputs = [out_f | out_b] @ Wd + bd   (M=16384, N=512, K=1024)
  {
    const int blocks = (1024 * 32 + 7) / 8;   // 4096
    gemm_wmma<<<blocks, 256, 0, stream>>>(O0, O1, WdS, bd, outp,
                                          16384, 32, 128, 128, Hsz, Hsz, 0, 1);
  }

  // 6) hidden = [h_f | h_b] @ Wd + bd  (final h lives in buffer p=0, T even)
  {
    float* hf = Hp;                       // dir 0, buf 0
    float* hb = Hp + (size_t)2 * 16384;   // dir 1, buf 0
    gemm_wmma<<<8, 256, 0, stream>>>(hf, hb, WdS, bd,
                                     outp + (size_t)Bsz * Tsz * Hsz,
                                     32, 32, 128, 128, Hsz, Hsz, 0, 0);
  }
}